// Circuit_84018150244642
// MI455X (gfx1250) — compile-verified
//
#include <hip/hip_runtime.h>
#include <stdint.h>

// ---------------------------------------------------------------------------
// MPS quantum-circuit forward pass for MI455X (gfx1250).
//
// Roofline: output = 84 MB -> 3.6 us at 23.3 TB/s HBM; arithmetic ~1 GFLOP.
// => bandwidth-bound. Strategy: one workgroup per batch element, full 80 KB
// state tensor resident in LDS (CDNA5: 320 KB/WGP), all gate algebra done in
// LDS restricted to the occupied 2^l x 2^l bond blocks, then a single 1-D
// Tensor-Data-Mover descriptor streams the finished state to HBM
// (tensor_store_from_lds + s_wait_tensorcnt). Theta (640 B) is prefetched
// with tensor_load_to_lds overlapped with the LDS zero-init.
// WMMA is deliberately NOT used: the only matmul is a real 2x2 (K=2) per
// bond cell on a memory-bound kernel -- TDM/LDS is the CDNA5 path that pays.
// ---------------------------------------------------------------------------

#define NQ     20
#define RK     16
#define NLAYER 4
#define PP     (NLAYER * 2 * NQ)   // 160 angles per batch element
#define BLOCK  256                 // 8 wave32 per workgroup

typedef unsigned int       u32;
typedef unsigned long long u64;

typedef unsigned int v4u __attribute__((ext_vector_type(4)));
typedef int          v4i __attribute__((ext_vector_type(4)));
typedef int          v8i __attribute__((ext_vector_type(8)));

#if __has_builtin(__builtin_amdgcn_tensor_load_to_lds) && \
    __has_builtin(__builtin_amdgcn_tensor_store_from_lds) && \
    __has_builtin(__builtin_amdgcn_s_wait_tensorcnt)
#define USE_TDM 1
#else
#define USE_TDM 0
#endif

struct __align__(16) Smem {
  float2 T[NQ][RK][RK][2];            // 81920 B; layout == (N,16,16,2) complex64
  float  th[PP];                      // staged angles
  float  ry_c[NQ], ry_s[NQ], rz_c[NQ], rz_s[NQ];
};

#if USE_TDM
// Pack a 1-D TDM descriptor (D#) per CDNA5 ISA ch.8:
// group0: [1:0]=count(1) valid, [63:32]=lds_addr, [120:64]=global_addr,
//         [127:126]=type(2).
// group1: [17:16]=data_size, [79:48]=tensor_dim0, [111:80]=tensor_dim1(=1),
//         [127:112]=tile_dim0, tile_dim1=0 (1-D), [207:160]=dim0_stride.
__device__ __forceinline__ void tdm_desc_1d(u32 lds_byte_addr, u64 gaddr,
                                            u32 n_elem, u32 elem_code,
                                            v4u& g0, v8i& g1) {
  union { u32 w[4]; v4u v; } a = {};
  union { u32 w[8]; v8i v; } b = {};
  a.w[0] = 1u;                                             // count=1, no gather
  a.w[1] = lds_byte_addr;                                  // lds_addr
  a.w[2] = (u32)(gaddr & 0xFFFFFFFFull);                   // global_addr[31:0]
  a.w[3] = (u32)((gaddr >> 32) & 0x1FFFFFFull)             // global_addr[56:32]
         | (2u << 30);                                     // type=2 ("image")
  b.w[0] = (elem_code & 3u) << 16;                         // data_size; mask=0
  b.w[1] = (n_elem & 0xFFFFu) << 16;                       // tensor_dim0[15:0]
  b.w[2] = ((n_elem >> 16) & 0xFFFFu) | (1u << 16);        // dim0 hi | tensor_dim1=1
  b.w[3] = (n_elem & 0xFFFFu) << 16;                       // tile_dim0 (<=65535)
  b.w[4] = 0;                                              // tile_dim1=0 -> 1-D
  b.w[5] = n_elem;                                         // tensor_dim0_stride lo
  b.w[6] = 0;
  b.w[7] = 0;
  g0 = a.v; g1 = b.v;
}

__device__ __forceinline__ void tdm_load_1d(u32 lds, u64 g, u32 n, u32 ec) {
  v4u g0; v8i g1; v4i z4 = {};
  tdm_desc_1d(lds, g, n, ec, g0, g1);
#if defined(__clang_major__) && (__clang_major__ >= 23)
  v8i z8 = {};
  __builtin_amdgcn_tensor_load_to_lds(g0, g1, z4, z4, z8, 0);
#else
  __builtin_amdgcn_tensor_load_to_lds(g0, g1, z4, z4, 0);
#endif
}

__device__ __forceinline__ void tdm_store_1d(u32 lds, u64 g, u32 n, u32 ec) {
  v4u g0; v8i g1; v4i z4 = {};
  tdm_desc_1d(lds, g, n, ec, g0, g1);
#if defined(__clang_major__) && (__clang_major__ >= 23)
  v8i z8 = {};
  __builtin_amdgcn_tensor_store_from_lds(g0, g1, z4, z4, z8, 0);
#else
  __builtin_amdgcn_tensor_store_from_lds(g0, g1, z4, z4, 0);
#endif
}
#endif  // USE_TDM

__global__ __launch_bounds__(BLOCK) void mps_circuit(
    const float* __restrict__ theta, float* __restrict__ out) {
  __shared__ Smem sm;
  const int tid = threadIdx.x;
  const int b   = blockIdx.x;

  // ---- stage theta for this batch element (TDM DMA, overlapped with init) --
#if USE_TDM
  if (tid == 0) {  // TDM ignores EXEC; issue once from wave 0
    tdm_load_1d((u32)(u64)&sm.th[0], (u64)(theta + (u64)b * PP),
                PP, 2u /*4-byte elems*/);
  }
#else
  if (tid < PP) sm.th[tid] = theta[(u64)b * PP + tid];
#endif

  // ---- zero-init the state while the DMA flies ----------------------------
  float4* tz = (float4*)&sm.T[0][0][0][0];
  constexpr int NF4 = (int)(sizeof(sm.T) / sizeof(float4));  // 5120
  for (int i = tid; i < NF4; i += BLOCK)
    tz[i] = make_float4(0.f, 0.f, 0.f, 0.f);

#if USE_TDM
  if (tid == 0) __builtin_amdgcn_s_wait_tensorcnt(0);
#endif
  __syncthreads();
  if (tid < NQ) sm.T[tid][0][0][0] = make_float2(1.f, 0.f);  // |0...0>
  __syncthreads();

  for (int l = 0; l < NLAYER; ++l) {
    const int d  = 1 << l;          // occupied bond dim at layer start
    const int dd = d * d;

    // per-qubit trig: RY half-angle (c,s) and RZ half-angle phase (zc,zs)
    if (tid < NQ) {
      float a = 0.5f * sm.th[l * 2 * NQ + tid];
      sm.ry_c[tid] = cosf(a); sm.ry_s[tid] = sinf(a);
      float z = 0.5f * sm.th[l * 2 * NQ + NQ + tid];
      sm.rz_c[tid] = cosf(z); sm.rz_s[tid] = sinf(z);
    }
    __syncthreads();

    // ---- RY+RZ fused, elementwise over occupied d x d block per site ------
    for (int it = tid; it < NQ * dd; it += BLOCK) {
      int q  = it / dd, c = it - q * dd;
      int le = c / d,   r = c - le * d;
      float2* cell = &sm.T[q][le][r][0];
      float2 x0 = cell[0], x1 = cell[1];
      float cc = sm.ry_c[q], ss = sm.ry_s[q];
      float zc = sm.rz_c[q], zs = sm.rz_s[q];
      float o0r = cc * x0.x - ss * x1.x, o0i = cc * x0.y - ss * x1.y;
      float o1r = ss * x0.x + cc * x1.x, o1i = ss * x0.y + cc * x1.y;
      // * exp(-i z) on phys 0, * exp(+i z) on phys 1
      cell[0] = make_float2(o0r * zc + o0i * zs, o0i * zc - o0r * zs);
      cell[1] = make_float2(o1r * zc - o1i * zs, o1i * zc + o1r * zs);
    }
    __syncthreads();

    // ---- CNOT ladder q -> q+1 (exact MPO; bond r maps to 2r+m) ------------
    // control site q: left dim already 2d (expanded by step q-1), right d.
    // target  site q+1: left d, right d. Cells outside are provably zero.
    const int Nc = 2 * dd;          // control source cells (2d x d)
    const int Nt = dd;              // target  source cells (d  x d)
    for (int q = 0; q < NQ - 1; ++q) {
      float2 a0 = make_float2(0.f, 0.f), a1 = make_float2(0.f, 0.f);
      int li = 0, ri = 0, role = 2;
      if (tid < Nc) {
        role = 0; li = tid / d; ri = tid - li * d;
        a0 = sm.T[q][li][ri][0]; a1 = sm.T[q][li][ri][1];
      } else if (tid < Nc + Nt) {
        role = 1; int t = tid - Nc; li = t / d; ri = t - li * d;
        a0 = sm.T[q + 1][li][ri][0]; a1 = sm.T[q + 1][li][ri][1];
      }
      __syncthreads();              // all reads done before in-place writes
      if (role == 0) {              // |0><0| (x) I  +  |1><1| (x) X, control
        sm.T[q][li][2 * ri    ][0] = a0;
        sm.T[q][li][2 * ri    ][1] = make_float2(0.f, 0.f);
        sm.T[q][li][2 * ri + 1][0] = make_float2(0.f, 0.f);
        sm.T[q][li][2 * ri + 1][1] = a1;
      } else if (role == 1) {       // target: m=0 identity, m=1 swaps phys
        sm.T[q + 1][2 * li    ][ri][0] = a0;
        sm.T[q + 1][2 * li    ][ri][1] = a1;
        sm.T[q + 1][2 * li + 1][ri][0] = a1;
        sm.T[q + 1][2 * li + 1][ri][1] = a0;
      }
      __syncthreads();              // writes visible before next step reads
    }
  }

  // ---- stream the finished 80 KB tensor to HBM ----------------------------
  // LDS layout of sm.T is exactly the required (N,16,16,2) complex64 order,
  // so one contiguous 1-D TDM store does the whole batch element.
#if USE_TDM
  if (tid == 0) {
    tdm_store_1d((u32)(u64)&sm.T[0][0][0][0],
                 (u64)out + (u64)b * sizeof(sm.T),
                 (u32)(sizeof(sm.T) / 8), 3u /*8-byte elems*/);
    __builtin_amdgcn_s_wait_tensorcnt(0);  // keep LDS alive until DMA drains
  }
#else
  float4* o4 = (float4*)out;
  for (int i = tid; i < NF4; i += BLOCK)
    o4[(u64)b * NF4 + i] = tz[i];
#endif
}

extern "C" void kernel_launch(void* const* d_in, const int* in_sizes, int n_in,
                              void* d_out, int out_size, void* d_ws,
                              size_t ws_size, hipStream_t stream) {
  (void)n_in; (void)out_size; (void)d_ws; (void)ws_size;
  const float* theta = (const float*)d_in[0];
  const int B = in_sizes[0] / PP;   // 1024 for the reference setup
  mps_circuit<<<dim3(B), dim3(BLOCK), 0, stream>>>(theta, (float*)d_out);
}